// GNNEncoder_32409823216438
// MI455X (gfx1250) — compile-verified
//
#include <hip/hip_runtime.h>

// ---------------------------------------------------------------------------
// GraphSAGE 2-layer encoder for MI455X (gfx1250, wave32).
// Memory/atomic-bound workload (~1.5 GB traffic -> ~65us at 23.3 TB/s);
// GEMMs (13 GFLOP total) routed through full-precision V_WMMA_F32_16X16X4_F32
// so numerics match the fp32 reference exactly while using the matrix pipe.
// ---------------------------------------------------------------------------

typedef __attribute__((ext_vector_type(2))) float v2f;
typedef __attribute__((ext_vector_type(8))) float v8f;

#define N_NODES 100000
#define D_FEAT  128
#define N_EDGES 640000

// ---------------------------------------------------------------------------
// Zero the accumulation workspace (agg + deg), float4 at a time.
// ---------------------------------------------------------------------------
__global__ void zero_f32(float* __restrict__ p, int n4) {
  int i = blockIdx.x * blockDim.x + threadIdx.x;
  if (i < n4) ((float4*)p)[i] = make_float4(0.f, 0.f, 0.f, 0.f);
}

// ---------------------------------------------------------------------------
// One wave32 per edge: lanes 0..31 each move a float4 (128 floats total).
// Coalesced 512B gather of x[src], coalesced atomic-add scatter onto agg[dst].
// Lane 0 counts the degree.
// ---------------------------------------------------------------------------
__global__ __launch_bounds__(256)
void scatter_mean_accum(const float* __restrict__ x,
                        const int* __restrict__ src,
                        const int* __restrict__ dst,
                        float* __restrict__ agg,
                        float* __restrict__ deg) {
  int wave = (int)((blockIdx.x * blockDim.x + threadIdx.x) >> 5);
  int lane = threadIdx.x & 31;
  if (wave >= N_EDGES) return;

  int s = src[wave];
  int d = dst[wave];

  const float4 v = ((const float4*)(x + (size_t)s * D_FEAT))[lane];
  float* a = agg + (size_t)d * D_FEAT + lane * 4;
  atomicAdd(a + 0, v.x);
  atomicAdd(a + 1, v.y);
  atomicAdd(a + 2, v.z);
  atomicAdd(a + 3, v.w);
  if (lane == 0) atomicAdd(deg + d, 1.0f);
}

// ---------------------------------------------------------------------------
// out[m,n] = relu?( (agg[m,:]/max(deg[m],1)) . w_l[n,:] + x[m,:] . w_r[n,:] + b_l[n] )
//
// One wave computes a 16x16 output tile via V_WMMA_F32_16X16X4_F32,
// accumulating K=128 (mean path) + K=128 (root path) in steps of 4.
// Block = 8 waves = 16 rows x 128 cols. Grid = N/16 (exact), EXEC all-1s.
//
// Fragment layout (ISA 7.12.2, f32):
//   A 16x4 : lane(li,hi) holds row m0+li, K = {k0+2*hi, k0+2*hi+1}  (v2f)
//   B 4x16 : lane(li,hi) holds col n0+li, K = {k0+2*hi, k0+2*hi+1}  (v2f)
//   C/D    : VGPR r -> row m0 + r + 8*hi, col n0+li                  (v8f)
// ---------------------------------------------------------------------------
__global__ __launch_bounds__(256)
void sage_layer(const float* __restrict__ xin,
                const float* __restrict__ agg,
                const float* __restrict__ deg,
                const float* __restrict__ w_l,   // [128,128] row-major
                const float* __restrict__ b_l,   // [128]
                const float* __restrict__ w_r,   // [128,128] row-major
                float* __restrict__ out,         // [N,128]
                int do_relu) {
  const int wave = threadIdx.x >> 5;   // 0..7 : column tile
  const int lane = threadIdx.x & 31;
  const int hi   = lane >> 4;          // 0 or 1
  const int li   = lane & 15;
  const int m0   = blockIdx.x * 16;
  const int n0   = wave * 16;

  const int rowA = m0 + li;
  const float dv = deg[rowA];
  const float rs = 1.0f / fmaxf(dv, 1.0f);   // 1/max(deg,1) for mean

  const float* aggRow = agg + (size_t)rowA * D_FEAT + 2 * hi;
  const float* xRow   = xin + (size_t)rowA * D_FEAT + 2 * hi;
  const float* wlRow  = w_l + (size_t)(n0 + li) * D_FEAT + 2 * hi;
  const float* wrRow  = w_r + (size_t)(n0 + li) * D_FEAT + 2 * hi;

  v8f acc = {};

  // mean-aggregated neighbors @ w_l^T  (scale applied on the A fragment)
  #pragma unroll 4
  for (int k0 = 0; k0 < D_FEAT; k0 += 4) {
    v2f a = *(const v2f*)(aggRow + k0);
    a = a * rs;
    v2f b = *(const v2f*)(wlRow + k0);
    acc = __builtin_amdgcn_wmma_f32_16x16x4_f32(
        /*neg_a=*/false, a, /*neg_b=*/false, b,
        /*c_mod=*/(short)0, acc, /*reuse_a=*/false, /*reuse_b=*/false);
  }

  // root features @ w_r^T
  #pragma unroll 4
  for (int k0 = 0; k0 < D_FEAT; k0 += 4) {
    v2f a = *(const v2f*)(xRow + k0);
    v2f b = *(const v2f*)(wrRow + k0);
    acc = __builtin_amdgcn_wmma_f32_16x16x4_f32(
        false, a, false, b, (short)0, acc, false, false);
  }

  const float bias = b_l[n0 + li];
  float* outBase = out + (size_t)(m0 + 8 * hi) * D_FEAT + n0 + li;
  #pragma unroll
  for (int r = 0; r < 8; ++r) {
    float v = acc[r] + bias;
    if (do_relu) v = fmaxf(v, 0.0f);
    outBase[(size_t)r * D_FEAT] = v;
  }
}

// ---------------------------------------------------------------------------
// Inputs (setup_inputs order):
//   0: x [100000,128] f32      4: w2_l [128,128] f32
//   1: w1_l [128,128] f32      5: b2_l [128] f32
//   2: b1_l [128] f32          6: w2_r [128,128] f32
//   3: w1_r [128,128] f32      7: edge_index [2,640000] int
// Workspace: agg (N*128 f32) | deg (N f32) | h (N*128 f32)  ~103 MB
// ---------------------------------------------------------------------------
extern "C" void kernel_launch(void* const* d_in, const int* in_sizes, int n_in,
                              void* d_out, int out_size, void* d_ws, size_t ws_size,
                              hipStream_t stream) {
  const float* x    = (const float*)d_in[0];
  const float* w1_l = (const float*)d_in[1];
  const float* b1_l = (const float*)d_in[2];
  const float* w1_r = (const float*)d_in[3];
  const float* w2_l = (const float*)d_in[4];
  const float* b2_l = (const float*)d_in[5];
  const float* w2_r = (const float*)d_in[6];
  const int*   edges = (const int*)d_in[7];
  const int*   src  = edges;
  const int*   dst  = edges + N_EDGES;

  float* agg = (float*)d_ws;
  float* deg = agg + (size_t)N_NODES * D_FEAT;
  float* h   = deg + N_NODES;
  float* out = (float*)d_out;

  const int zeroN  = N_NODES * D_FEAT + N_NODES;  // 12,900,000 floats (mult of 4)
  const int zeroN4 = zeroN / 4;
  dim3 zb(256), zg((zeroN4 + 255) / 256);
  dim3 sb(256), sg((N_EDGES * 32 + 255) / 256);   // one wave32 per edge
  dim3 lb(256), lg(N_NODES / 16);                 // 6250 blocks, exact

  // ---- layer 1 ----
  zero_f32<<<zg, zb, 0, stream>>>(agg, zeroN4);
  scatter_mean_accum<<<sg, sb, 0, stream>>>(x, src, dst, agg, deg);
  sage_layer<<<lg, lb, 0, stream>>>(x, agg, deg, w1_l, b1_l, w1_r, h, 1);

  // ---- layer 2 ----
  zero_f32<<<zg, zb, 0, stream>>>(agg, zeroN4);
  scatter_mean_accum<<<sg, sb, 0, stream>>>(h, src, dst, agg, deg);
  sage_layer<<<lg, lb, 0, stream>>>(h, agg, deg, w2_l, b2_l, w2_r, out, 0);
}